// EntityAwareAttention_33981781246252
// MI455X (gfx1250) — compile-verified
//
#include <hip/hip_runtime.h>
#include <hip/hip_bf16.h>
#include <math.h>
#include <stdint.h>

// Problem constants
#define BB 64
#define LL 256
#define HH 1024
#define PP 64
#define AA 512
#define TT 8
#define FF (HH + 2 * PP)   // 1152

typedef __attribute__((ext_vector_type(2))) float v2f;
typedef __attribute__((ext_vector_type(8))) float v8f;
typedef __attribute__((ext_vector_type(4))) int v4i;

#define AS3 __attribute__((address_space(3)))

#if __has_builtin(__builtin_amdgcn_global_load_async_to_lds_b128)
#define HAS_ASYNC_LDS 1
#else
#define HAS_ASYNC_LDS 0
#endif

// Copy 16B global -> LDS. Async (DMA, ASYNCcnt) when available, else VALU.
__device__ __forceinline__ void cp_b128(const float* __restrict__ g, float* l) {
#if HAS_ASYNC_LDS
  // param0: generic int4* (per clang diagnostic); param1: LDS int4*.
  // Low 32 bits of a generic LDS address == LDS byte offset (flat aperture).
  __builtin_amdgcn_global_load_async_to_lds_b128(
      (v4i*)g,
      (AS3 v4i*)(uint32_t)(uintptr_t)l, 0, 0);
#else
  const float4 v = *(const float4*)g;
  l[0] = v.x; l[1] = v.y; l[2] = v.z; l[3] = v.w;
#endif
}

__device__ __forceinline__ void wait_async0() {
#if HAS_ASYNC_LDS
#if __has_builtin(__builtin_amdgcn_s_wait_asynccnt)
  __builtin_amdgcn_s_wait_asynccnt(0);
#else
  asm volatile("s_wait_asynccnt 0" ::: "memory");
#endif
#endif
}

// ---------------------------------------------------------------------------
// Kernel 1: gather entity hiddens + latent type (softmax over T=8 types)
// ent_feat[b] = [e1_h | e1_type | e2_h | e2_type]  (4*H floats)
// ---------------------------------------------------------------------------
__global__ __launch_bounds__(256) void entity_kernel(
    const float* __restrict__ wh, const int* __restrict__ e1_end,
    const int* __restrict__ e2_end, const float* __restrict__ type_emb,
    float* __restrict__ ent_feat) {
  const int b = blockIdx.x;
  const int t = threadIdx.x;           // 256 threads = 8 waves
  const int wave = t >> 5, lane = t & 31;

  __shared__ float red[TT][8];
  __shared__ float alpha_s[TT];

  for (int ent = 0; ent < 2; ++ent) {
    const int epos = (ent == 0) ? e1_end[b] : e2_end[b];
    const float* eh = wh + ((size_t)b * LL + epos) * HH;

    float part[TT];
#pragma unroll
    for (int ty = 0; ty < TT; ++ty) part[ty] = 0.f;
    for (int h = t; h < HH; h += 256) {
      const float x = eh[h];
#pragma unroll
      for (int ty = 0; ty < TT; ++ty) part[ty] += x * type_emb[ty * HH + h];
    }
#pragma unroll
    for (int off = 16; off > 0; off >>= 1)
#pragma unroll
      for (int ty = 0; ty < TT; ++ty) part[ty] += __shfl_down(part[ty], off, 32);
    if (lane == 0)
#pragma unroll
      for (int ty = 0; ty < TT; ++ty) red[ty][wave] = part[ty];
    __syncthreads();

    if (t == 0) {
      float sc[TT];
      float mx = -1e30f;
#pragma unroll
      for (int ty = 0; ty < TT; ++ty) {
        float s = 0.f;
#pragma unroll
        for (int w = 0; w < 8; ++w) s += red[ty][w];
        sc[ty] = s;
        mx = fmaxf(mx, s);
      }
      float sum = 0.f;
#pragma unroll
      for (int ty = 0; ty < TT; ++ty) { sc[ty] = __expf(sc[ty] - mx); sum += sc[ty]; }
      const float inv = 1.f / sum;
#pragma unroll
      for (int ty = 0; ty < TT; ++ty) alpha_s[ty] = sc[ty] * inv;
    }
    __syncthreads();

    float a_[TT];
#pragma unroll
    for (int ty = 0; ty < TT; ++ty) a_[ty] = alpha_s[ty];

    float* dst_h = ent_feat + (size_t)b * 4 * HH + ent * 2 * HH;
    float* dst_t = dst_h + HH;
    for (int h = t; h < HH; h += 256) {
      const float x = eh[h];
      float lt = 0.f;
#pragma unroll
      for (int ty = 0; ty < TT; ++ty) lt += a_[ty] * type_emb[ty * HH + h];
      dst_h[h] = x;
      dst_t[h] = lt;
    }
    __syncthreads();
  }
}

// ---------------------------------------------------------------------------
// Kernel 2: dense_ent[64,512] = ent_feat[64,4096] @ W_ent[4096,512]
// WMMA f32 16x16x4.  Grid: 32 blocks (N tiles) x 128 threads (4 waves = 4 M tiles)
// ---------------------------------------------------------------------------
__global__ __launch_bounds__(128) void dense_ent_kernel(
    const float* __restrict__ ent_feat, const float* __restrict__ W_ent,
    float* __restrict__ dense_ent) {
  const int ntile = blockIdx.x;            // 0..31
  const int wave = threadIdx.x >> 5;       // 0..3  == mtile
  const int lane = threadIdx.x & 31;
  const int m = lane & 15, kh = lane >> 4;

  const float* arow = ent_feat + (size_t)(wave * 16 + m) * (4 * HH) + 2 * kh;
  const int n = ntile * 16 + m;

  v8f acc = {};
#pragma unroll 4
  for (int k = 0; k < 4 * HH; k += 4) {
    v2f a;
    a.x = arow[k];
    a.y = arow[k + 1];
    v2f bb;
    bb.x = W_ent[(size_t)(k + 2 * kh) * AA + n];
    bb.y = W_ent[(size_t)(k + 2 * kh + 1) * AA + n];
    acc = __builtin_amdgcn_wmma_f32_16x16x4_f32(false, a, false, bb, (short)0,
                                                acc, false, false);
  }
#pragma unroll
  for (int vv = 0; vv < 8; ++vv) {
    const int row = wave * 16 + vv + 8 * kh;   // C/D layout: M = v + 8*(lane>=16)
    dense_ent[(size_t)row * AA + n] = acc[vv];
  }
}

// ---------------------------------------------------------------------------
// Kernel 3 (main): fused dense_pos (WMMA) + bias + tanh + dot(v) -> vu[B,L]
// Workgroup = 8 waves, computes a 64(L) x 512(A) tile of one batch.
//   wave mg = wave>>2 (M tiles {2mg,2mg+1}), ng = wave&3 (N tiles ng*8..+7)
// K loop over F=1152 in chunks of 8, double-buffered async global->LDS DMA.
// ---------------------------------------------------------------------------
#define KCH 8
#define LDA 12    // ldsA row stride (floats): even (b64 align), conflict-free
#define LDB 520   // ldsB row stride (floats): mult of 4, half-wave bank split
#define NCH (FF / KCH)   // 144 chunks

__global__ __launch_bounds__(256) void attn_kernel(
    const float* __restrict__ wh, const float* __restrict__ pe1,
    const float* __restrict__ pe2, const float* __restrict__ W_pos,
    const float* __restrict__ dense_ent, const float* __restrict__ vvec,
    float* __restrict__ vu) {
  const int b = blockIdx.y;
  const int l0 = blockIdx.x * 64;
  const int tid = threadIdx.x;
  const int wave = tid >> 5, lane = tid & 31;
  const int mg = wave >> 2;   // 0..1 -> M tiles {2mg, 2mg+1}
  const int ng = wave & 3;    // 0..3 -> N tiles ng*8 .. ng*8+7
  const int m = lane & 15, kh = lane >> 4;

  __shared__ float ldsA[2][64 * LDA];
  __shared__ float ldsB[2][KCH * LDB];
  __shared__ float vu_lds[64];

  v8f acc[2][8] = {};

  // preload v fragment (col n = (ng*8+j)*16 + m)
  float vfrag[8];
#pragma unroll
  for (int j = 0; j < 8; ++j) vfrag[j] = vvec[(ng * 8 + j) * 16 + m];

  // issue async loads for chunk starting at k0 into buffer `buf`
  auto load_chunk = [&](int k0, int buf) {
    const float* src;
    int stride, koff;
    if (k0 < HH) {
      src = wh + ((size_t)b * LL + l0) * HH;  stride = HH;  koff = k0;
    } else if (k0 < HH + PP) {
      src = pe1 + ((size_t)b * LL + l0) * PP; stride = PP;  koff = k0 - HH;
    } else {
      src = pe2 + ((size_t)b * LL + l0) * PP; stride = PP;  koff = k0 - HH - PP;
    }
    // A tile: 64 rows x 8 floats -> 128 b128 transfers (threads 0..127)
    if (tid < 128) {
      const int r = tid >> 1, q = tid & 1;
      cp_b128(src + (size_t)r * stride + koff + q * 4, &ldsA[buf][r * LDA + q * 4]);
    }
    // B tile: 8 rows x 512 floats -> 1024 b128 transfers (4 per thread)
#pragma unroll
    for (int i = 0; i < 4; ++i) {
      const int q = tid + 256 * i;            // 0..1023 quads
      const int r = q >> 7, c = (q & 127) * 4;
      cp_b128(W_pos + (size_t)(k0 + r) * AA + c, &ldsB[buf][r * LDB + c]);
    }
  };

  load_chunk(0, 0);
  for (int c = 0; c < NCH; ++c) {
    const int cur = c & 1;
    wait_async0();       // this wave's DMA for chunk c has landed in LDS
    __syncthreads();     // all waves done: chunk c ready, buf cur^1 consumed

    if (c + 1 < NCH) load_chunk((c + 1) * KCH, cur ^ 1);  // overlap next DMA

    // compute on buffer `cur`: 2 K-steps of 4, 16 WMMAs each
#pragma unroll
    for (int ks = 0; ks < KCH / 4; ++ks) {
      const int kk = ks * 4 + 2 * kh;      // A lanes 0-15: K0/K1, 16-31: K2/K3
      v2f aF[2];
#pragma unroll
      for (int i = 0; i < 2; ++i) {
        const float* ap = &ldsA[cur][((mg * 2 + i) * 16 + m) * LDA + kk];
        aF[i].x = ap[0];
        aF[i].y = ap[1];
      }
#pragma unroll
      for (int j = 0; j < 8; ++j) {
        const int n = (ng * 8 + j) * 16 + m;
        v2f bF;
        bF.x = ldsB[cur][kk * LDB + n];
        bF.y = ldsB[cur][(kk + 1) * LDB + n];
        acc[0][j] = __builtin_amdgcn_wmma_f32_16x16x4_f32(
            false, aF[0], false, bF, (short)0, acc[0][j], false, false);
        acc[1][j] = __builtin_amdgcn_wmma_f32_16x16x4_f32(
            false, aF[1], false, bF, (short)0, acc[1][j], false, false);
      }
    }
  }

  // epilogue: bias (interleaved repeat == dense_ent[b, 2l + (a>=256)]),
  // tanh, dot with v, reduce rows across lanes/waves via ds_add_f32
  if (tid < 64) vu_lds[tid] = 0.f;
  __syncthreads();

  const float* de = dense_ent + (size_t)b * AA;
#pragma unroll
  for (int i = 0; i < 2; ++i) {
#pragma unroll
    for (int vv = 0; vv < 8; ++vv) {
      const int lloc = (mg * 2 + i) * 16 + vv + 8 * kh;  // row in [0,64)
      const int l = l0 + lloc;
      const float de0 = de[2 * l];
      const float de1 = de[2 * l + 1];
      float s = 0.f;
#pragma unroll
      for (int j = 0; j < 8; ++j) {
        const int ntile = ng * 8 + j;                    // a>=256 <=> ntile>=16
        const float bias = (ntile >= 16) ? de1 : de0;
        const float u = tanhf(acc[i][j][vv] + bias);
        s += u * vfrag[j];
      }
      atomicAdd(&vu_lds[lloc], s);                       // ds_add_f32
    }
  }
  __syncthreads();
  if (tid < 64) vu[(size_t)b * LL + l0 + tid] = vu_lds[tid];
}

// ---------------------------------------------------------------------------
// Kernel 4: softmax over L per batch + z = sum_l alpha[l] * wh[b,l,:]
// ---------------------------------------------------------------------------
__global__ __launch_bounds__(256) void softmax_z_kernel(
    const float* __restrict__ wh, const float* __restrict__ vu,
    float* __restrict__ z) {
  const int b = blockIdx.x;
  const int t = threadIdx.x;        // 256 threads; one vu element each
  const int wave = t >> 5, lane = t & 31;
  __shared__ float alpha[LL];
  __shared__ float red[8];

  const float x = vu[(size_t)b * LL + t];

  float mx = x;
#pragma unroll
  for (int off = 16; off > 0; off >>= 1) mx = fmaxf(mx, __shfl_xor(mx, off, 32));
  if (lane == 0) red[wave] = mx;
  __syncthreads();
  mx = red[0];
#pragma unroll
  for (int w = 1; w < 8; ++w) mx = fmaxf(mx, red[w]);
  __syncthreads();

  const float e = __expf(x - mx);
  float s = e;
#pragma unroll
  for (int off = 16; off > 0; off >>= 1) s += __shfl_xor(s, off, 32);
  if (lane == 0) red[wave] = s;
  __syncthreads();
  s = 0.f;
#pragma unroll
  for (int w = 0; w < 8; ++w) s += red[w];
  alpha[t] = e / s;
  __syncthreads();

  for (int h = t; h < HH; h += 256) {
    const float* col = wh + (size_t)b * LL * HH + h;
    float acc = 0.f;
#pragma unroll 4
    for (int l = 0; l < LL; ++l) acc += alpha[l] * col[(size_t)l * HH];
    z[(size_t)b * HH + h] = acc;
  }
}

// ---------------------------------------------------------------------------
extern "C" void kernel_launch(void* const* d_in, const int* in_sizes, int n_in,
                              void* d_out, int out_size, void* d_ws, size_t ws_size,
                              hipStream_t stream) {
  const float* wh   = (const float*)d_in[0];
  const float* pe1  = (const float*)d_in[1];
  const float* pe2  = (const float*)d_in[2];
  const int*   e1e  = (const int*)d_in[3];
  const int*   e2e  = (const int*)d_in[4];
  const float* temb = (const float*)d_in[5];
  const float* Wp   = (const float*)d_in[6];
  const float* We   = (const float*)d_in[7];
  const float* vv   = (const float*)d_in[8];
  float* z = (float*)d_out;

  float* ent_feat  = (float*)d_ws;                     // 64*4096 floats
  float* dense_ent = ent_feat + (size_t)BB * 4 * HH;   // 64*512 floats
  float* vu        = dense_ent + (size_t)BB * AA;      // 64*256 floats

  entity_kernel<<<BB, 256, 0, stream>>>(wh, e1e, e2e, temb, ent_feat);
  dense_ent_kernel<<<AA / 16, 128, 0, stream>>>(ent_feat, We, dense_ent);
  attn_kernel<<<dim3(LL / 64, BB), 256, 0, stream>>>(wh, pe1, pe2, Wp, dense_ent,
                                                     vv, vu);
  softmax_z_kernel<<<BB, 256, 0, stream>>>(wh, vu, z);
}